// _FlaxMPS_69166153335041
// MI455X (gfx1250) — compile-verified
//
#include <hip/hip_runtime.h>
#include <hip/hip_bf16.h>
#include <math.h>

#define D       128
#define DP      132          // padded row stride (floats): 4*row%64 -> conflict-free frags, 8B-aligned rows
#define NSITES  64
#define BATCHSZ 128
#define TPB     256

typedef __attribute__((ext_vector_type(2))) float v2f;
typedef __attribute__((ext_vector_type(8))) float v8f;
typedef __attribute__((ext_vector_type(4))) int   v4i;

#ifndef __has_builtin
#define __has_builtin(x) 0
#endif

#if __has_builtin(__builtin_amdgcn_global_load_async_to_lds_b128)
#define USE_ASYNC 1
#else
#define USE_ASYNC 0
#endif

#define AS1 __attribute__((address_space(1)))
#define AS3 __attribute__((address_space(3)))

#if USE_ASYNC
static __device__ inline void wait_async0() {
#if __has_builtin(__builtin_amdgcn_s_wait_asynccnt)
  __builtin_amdgcn_s_wait_asynccnt(0);
#else
  asm volatile("s_wait_asynccnt 0" ::: "memory");
#endif
}
#endif

static __device__ inline v8f wmma4(v2f a, v2f b, v8f c) {
  // D(16x16,f32) = A(16x4,f32) * B(4x16,f32) + C
  return __builtin_amdgcn_wmma_f32_16x16x4_f32(
      /*neg_a=*/false, a, /*neg_b=*/false, b,
      /*c_mod=*/(short)0, c, /*reuse_a=*/false, /*reuse_b=*/false);
}

struct SmemT {
  float  Cre[D * DP];     // carry real plane
  float  Cim[D * DP];     // carry imag plane
  float2 Abuf[D * D];     // current site matrix, interleaved complex
  float  tr_re[D];
  float  tr_im[D];
};

__global__ __launch_bounds__(TPB) void mps_chain_kernel(
    const int* __restrict__ s, const float2* __restrict__ A,
    float* __restrict__ out)
{
  __shared__ SmemT sm;
  const int tid   = threadIdx.x;
  const int b     = blockIdx.x;
  const int lane  = tid & 31;
  const int wid   = tid >> 5;        // 8 waves
  const int nlane = lane & 15;
  const int klane = lane >> 4;       // 0 or 1 (lane half)
  const int mbase = wid * 16;        // wave's 16-row strip of C

  // ---- init carry = A[s[0]]  (I * A_s0), deinterleave into planes ----
  {
    int s0 = s[b * NSITES];
    s0 = s0 < 0 ? 0 : (s0 > 1 ? 1 : s0);
    const float2* A0 = A + (size_t)s0 * D * D;
    for (int idx = tid; idx < D * D; idx += TPB) {
      float2 v = A0[idx];
      int i = idx >> 7, j = idx & (D - 1);
      sm.Cre[i * DP + j] = v.x;
      sm.Cim[i * DP + j] = v.y;
    }
  }

  for (int t = 1; t < NSITES; ++t) {
    int st = s[b * NSITES + t];
    st = st < 0 ? 0 : (st > 1 ? 1 : st);
    const float2* At = A + (size_t)st * D * D;

    // ---- stage A[s_t] (128KB) into LDS via async-to-LDS ----
#if USE_ASYNC
    #pragma unroll 4
    for (int i = 0; i < (D * D) / (TPB * 2); ++i) {   // 32 iters, 16B/lane each
      int e = (tid + i * TPB) * 2;                    // complex-pair index
      __builtin_amdgcn_global_load_async_to_lds_b128(
          (AS1 v4i*)(At + e), (AS3 v4i*)(&sm.Abuf[e]), 0, 0);
    }
    wait_async0();
#else
    {
      const float4* Ag4 = reinterpret_cast<const float4*>(At);
      float4*       Al4 = reinterpret_cast<float4*>(sm.Abuf);
      for (int i = tid; i < (D * D) / 2; i += TPB) Al4[i] = Ag4[i];
    }
#endif
    __syncthreads();   // Abuf ready; carry planes ready from previous site

    // ---- acc = C * A_site, 3M complex multiply: 3 WMMAs per k-block/tile ----
    // P1 = Re(C)*Re(A), P2 = Im(C)*Im(A), P3 = (Re+Im)(C) * (Re+Im)(A)
    // Re = P1 - P2 ; Im = P3 - P1 - P2   (combined once per site)
    v8f p1[8], p2[8], p3[8];
    {
      v8f z = {};
      #pragma unroll
      for (int nt = 0; nt < 8; ++nt) { p1[nt] = z; p2[nt] = z; p3[nt] = z; }
    }

    const int arow = mbase + nlane;  // A-fragment: lane -> row M
    // Keep rolled: one iteration already has 24 independent WMMAs; unrolling
    // pushes VGPR allocation past 256 and triggers s_set_vgpr_msb churn.
    #pragma unroll 1
    for (int kb = 0; kb < D / 4; ++kb) {
      const int kk = kb * 4 + klane * 2;   // ISA A-layout: lanes>=16 hold K+2
      v2f are, aim, asum;
      are.x = sm.Cre[arow * DP + kk];
      are.y = sm.Cre[arow * DP + kk + 1];
      aim.x = sm.Cim[arow * DP + kk];
      aim.y = sm.Cim[arow * DP + kk + 1];
      asum.x = are.x + aim.x;
      asum.y = are.y + aim.y;

      #pragma unroll
      for (int nt = 0; nt < 8; ++nt) {
        const int col = nt * 16 + nlane;   // B-fragment: lane -> column N
        float2 e0 = sm.Abuf[kk * D + col];        // row K   (re,im)
        float2 e1 = sm.Abuf[(kk + 1) * D + col];  // row K+1 (re,im)
        v2f bre, bim, bsum;
        bre.x = e0.x; bre.y = e1.x;
        bim.x = e0.y; bim.y = e1.y;
        bsum.x = bre.x + bim.x;
        bsum.y = bre.y + bim.y;
        p1[nt] = wmma4(are,  bre,  p1[nt]);
        p2[nt] = wmma4(aim,  bim,  p2[nt]);
        p3[nt] = wmma4(asum, bsum, p3[nt]);
      }
    }
    __syncthreads();   // everyone done reading carry planes

    // ---- combine 3M partials and write D-fragments back into carry planes ----
    #pragma unroll
    for (int nt = 0; nt < 8; ++nt) {
      const int c = nt * 16 + nlane;
      #pragma unroll
      for (int v = 0; v < 8; ++v) {
        const int r = mbase + v + klane * 8;  // C/D layout: VGPR v -> row v (+8 upper half)
        float a = p1[nt][v], bb = p2[nt][v];
        sm.Cre[r * DP + c] = a - bb;
        sm.Cim[r * DP + c] = p3[nt][v] - a - bb;
      }
    }
    __syncthreads();
  }

  // ---- trace + complex log ----
  if (tid < D) {
    sm.tr_re[tid] = sm.Cre[tid * DP + tid];
    sm.tr_im[tid] = sm.Cim[tid * DP + tid];
  }
  __syncthreads();
  if (tid == 0) {
    float re = 0.f, im = 0.f;
    for (int i = 0; i < D; ++i) { re += sm.tr_re[i]; im += sm.tr_im[i]; }
    out[2 * b + 0] = 0.5f * logf(re * re + im * im);  // ln|z|
    out[2 * b + 1] = atan2f(im, re);                  // arg(z)
  }
}

extern "C" void kernel_launch(void* const* d_in, const int* in_sizes, int n_in,
                              void* d_out, int out_size, void* d_ws, size_t ws_size,
                              hipStream_t stream) {
  (void)in_sizes; (void)n_in; (void)d_ws; (void)ws_size; (void)out_size;
  const int*    s = (const int*)d_in[0];
  const float2* A = (const float2*)d_in[1];   // complex64 -> float2
  float*      out = (float*)d_out;            // complex64 out, interleaved re/im
  mps_chain_kernel<<<dim3(BATCHSZ), dim3(TPB), 0, stream>>>(s, A, out);
}